// StandGCN1_41532333752789
// MI455X (gfx1250) — compile-verified
//
#include <hip/hip_runtime.h>

// ---------------- problem constants (match reference) ----------------
#define N_NODES 100000
#define N_EDGES 3200000
#define NFEAT   512
#define NCLASS  40

typedef float v2f __attribute__((ext_vector_type(2)));
typedef float v8f __attribute__((ext_vector_type(8)));

// =====================================================================
// Kernel 1: xw[N,40] = x[N,512] @ W[40,512]^T  via V_WMMA_F32_16X16X4_F32
// One wave computes a 16(M) x 48(N, 40 valid) tile, looping K in steps of 4.
// A fragment (16x4 f32, 2 VGPRs): lane = M + 16*(K>>1); v0=K_even, v1=K_odd.
// B fragment (4x16 f32, 2 VGPRs): lane = N + 16*(K>>1) (mirror of A).
// D (16x16 f32, 8 VGPRs): vgpr i -> M = i + 8*(lane>>4), N = lane&15.
// =====================================================================
__global__ __launch_bounds__(256) void gcn_xw_wmma_kernel(
    const float* __restrict__ x, const float* __restrict__ W,
    float* __restrict__ xw) {
  const int wave = threadIdx.x >> 5;
  const int lane = threadIdx.x & 31;
  const int rowbase = (blockIdx.x * 8 + wave) * 16;
  if (rowbase >= N_NODES) return;  // wave-uniform: EXEC stays all-1s for WMMA

  const int m  = lane & 15;   // M (for A) / N (for B) within tile
  const int kh = lane >> 4;   // K half-select

  // A source: 16 rows of x, this lane's K pair
  const float* xr = x + (size_t)(rowbase + m) * NFEAT + 2 * kh;

  // B source: W is [NCLASS, NFEAT]; B[k][n] = W[n][k]. Three 16-col tiles.
  const int n0 = m;
  const int n1 = 16 + m;
  const int n2 = 32 + m;                       // 32..47; >=40 invalid
  const int n2c = (n2 < NCLASS) ? n2 : (n2 - NCLASS);  // clamp to a valid row,
                                                       // junk cols never stored
  const float* w0 = W + (size_t)n0  * NFEAT + 2 * kh;
  const float* w1 = W + (size_t)n1  * NFEAT + 2 * kh;
  const float* w2 = W + (size_t)n2c * NFEAT + 2 * kh;

  v8f c0 = {}; v8f c1 = {}; v8f c2 = {};

  for (int k0 = 0; k0 < NFEAT; k0 += 4) {
    v2f a  = *reinterpret_cast<const v2f*>(xr + k0);
    v2f b0 = *reinterpret_cast<const v2f*>(w0 + k0);
    v2f b1 = *reinterpret_cast<const v2f*>(w1 + k0);
    v2f b2 = *reinterpret_cast<const v2f*>(w2 + k0);
    // 8-arg form: (neg_a, A, neg_b, B, c_mod, C, reuse_a, reuse_b)
    c0 = __builtin_amdgcn_wmma_f32_16x16x4_f32(false, a, false, b0, (short)0, c0, false, false);
    c1 = __builtin_amdgcn_wmma_f32_16x16x4_f32(false, a, false, b1, (short)0, c1, false, false);
    c2 = __builtin_amdgcn_wmma_f32_16x16x4_f32(false, a, false, b2, (short)0, c2, false, false);
  }

  // Store D tiles (cols >= 40 discarded)
  const int col0 = m, col1 = 16 + m, col2 = 32 + m;
#pragma unroll
  for (int i = 0; i < 8; ++i) {
    const int row = rowbase + i + 8 * kh;
    float* o = xw + (size_t)row * NCLASS;
    o[col0] = c0[i];
    o[col1] = c1[i];
    if (col2 < NCLASS) o[col2] = c2[i];
  }
}

// =====================================================================
// Kernel 2a: deg[n] = 1.0 (appended self-loop weight)
// =====================================================================
__global__ void gcn_deg_init_kernel(float* __restrict__ deg) {
  const int n = blockIdx.x * blockDim.x + threadIdx.x;
  if (n < N_NODES) deg[n] = 1.0f;
}

// =====================================================================
// Kernel 2b: deg[col] += 1 for every non-self-loop edge (exact: < 2^24)
// =====================================================================
__global__ void gcn_deg_count_kernel(const long long* __restrict__ adj,
                                     float* __restrict__ deg) {
  const int e = blockIdx.x * blockDim.x + threadIdx.x;
  if (e >= N_EDGES) return;
  const long long r = adj[e];
  const long long c = adj[(size_t)N_EDGES + e];
  if (r != c) atomicAdd(&deg[(int)c], 1.0f);
}

// =====================================================================
// Kernel 2c: deg[n] -> 1/deg[n]   (norm = dis[col]^2 * w = w / deg[col])
// =====================================================================
__global__ void gcn_deg_inv_kernel(float* __restrict__ deg) {
  const int n = blockIdx.x * blockDim.x + threadIdx.x;
  if (n < N_NODES) deg[n] = 1.0f / deg[n];  // deg >= 1 always
}

// =====================================================================
// Kernel 3: out[n][c] = b[c] + xw[n][c] * invdeg[n]   (self-loop + bias,
// and full initialization of d_out each call)
// =====================================================================
__global__ void gcn_out_init_kernel(const float* __restrict__ xw,
                                    const float* __restrict__ invdeg,
                                    const float* __restrict__ b,
                                    float* __restrict__ out) {
  const int t = blockIdx.x * blockDim.x + threadIdx.x;
  if (t >= N_NODES * NCLASS) return;
  const int n = t / NCLASS;
  const int c = t - n * NCLASS;
  out[t] = b[c] + xw[t] * invdeg[n];
}

// =====================================================================
// Kernel 4: edge scatter. One wave32 per edge; lane = column.
// out[col][j] += xw[row][j] * invdeg[col]  for j in [0,40)
// Coalesced 128B gathers/atomics; xw & out are L2-resident (16 MB each).
// =====================================================================
__global__ __launch_bounds__(256) void gcn_scatter_kernel(
    const long long* __restrict__ adj, const float* __restrict__ xw,
    const float* __restrict__ invdeg, float* __restrict__ out) {
  const long long tid = (long long)blockIdx.x * blockDim.x + threadIdx.x;
  const long long e  = tid >> 5;
  const int lane     = (int)(tid & 31);
  if (e >= N_EDGES) return;
  const int r = (int)adj[e];
  const int c = (int)adj[(size_t)N_EDGES + e];
  if (r == c) return;  // self-loops dropped (weight 0)
  const float inv = invdeg[c];
  const float* src = xw + (size_t)r * NCLASS;
  float* dst = out + (size_t)c * NCLASS;
  atomicAdd(&dst[lane], src[lane] * inv);            // cols 0..31
  if (lane < NCLASS - 32) {                          // cols 32..39
    atomicAdd(&dst[32 + lane], src[32 + lane] * inv);
  }
}

// =====================================================================
extern "C" void kernel_launch(void* const* d_in, const int* in_sizes, int n_in,
                              void* d_out, int out_size, void* d_ws, size_t ws_size,
                              hipStream_t stream) {
  const float*     x   = (const float*)d_in[0];      // [N, 512] f32
  const long long* adj = (const long long*)d_in[1];  // [2, E] i64
  const float*     W   = (const float*)d_in[2];      // [40, 512] f32
  const float*     b   = (const float*)d_in[3];      // [40] f32
  float*           out = (float*)d_out;              // [N, 40] f32

  // workspace layout
  float* xw  = (float*)d_ws;                                       // 16.0 MB
  float* deg = (float*)((char*)d_ws +
                        (size_t)N_NODES * NCLASS * sizeof(float)); // 0.4 MB

  // 1) GEMM via f32 WMMA: 6250 waves of 16 rows, 8 waves/block
  {
    const int nwaves = (N_NODES + 15) / 16;
    const int blocks = (nwaves + 7) / 8;
    gcn_xw_wmma_kernel<<<blocks, 256, 0, stream>>>(x, W, xw);
  }
  // 2) degree (with self-loop), then invert
  gcn_deg_init_kernel<<<(N_NODES + 255) / 256, 256, 0, stream>>>(deg);
  gcn_deg_count_kernel<<<(N_EDGES + 255) / 256, 256, 0, stream>>>(adj, deg);
  gcn_deg_inv_kernel<<<(N_NODES + 255) / 256, 256, 0, stream>>>(deg);
  // 3) self-loop contribution + bias (initializes out)
  gcn_out_init_kernel<<<(N_NODES * NCLASS + 255) / 256, 256, 0, stream>>>(
      xw, deg, b, out);
  // 4) edge aggregation: one wave per edge
  {
    const long long threads = (long long)N_EDGES * 32;
    const int blocks = (int)((threads + 255) / 256);
    gcn_scatter_kernel<<<blocks, 256, 0, stream>>>(adj, xw, deg, out);
  }
}